// PairEmbedFull_36318243455342
// MI455X (gfx1250) — compile-verified
//
#include <hip/hip_runtime.h>

#define EPS_F 1e-8f
#define BN_EPS_F 1e-5f

typedef float v2f __attribute__((ext_vector_type(2)));
typedef float v8f __attribute__((ext_vector_type(8)));

__device__ __forceinline__ v8f wmma4(v2f a, v2f b, v8f c) {
  // D(16x16) = A(16x4) * B(4x16) + C, full f32, wave32
  return __builtin_amdgcn_wmma_f32_16x16x4_f32(false, a, false, b, 0, c, false, false);
}

__device__ __forceinline__ float gelu_exact(float v) {
  return 0.5f * v * (1.0f + erff(v * 0.70710678118654752f));
}

// invert p = i*(i+1)/2 + j  (np.tril_indices order)
__device__ __forceinline__ void decode_ij(int p, int& i, int& j) {
  int t = (int)((sqrtf(8.0f * (float)p + 1.0f) - 1.0f) * 0.5f);
  while ((t + 1) * (t + 2) / 2 <= p) ++t;
  while (t * (t + 1) / 2 > p) --t;
  i = t;
  j = p - t * (t + 1) / 2;
}

// One 16-row (pairs) x 64-col (channels) GEMM via 16x16x4 f32 WMMA.
// st: LDS activation tile [16][68] (row = pair, col = channel/K)
// W : LDS weight matrix   [64][wstride], row = out-channel n, col = k
__device__ __forceinline__ void gemm_tile(const float* st, const float* W,
                                          int wstride, int K, int row, int khalf,
                                          v8f acc[4]) {
  for (int k0 = 0; k0 < K; k0 += 4) {
    const int ka = k0 + khalf;               // lanes 0-15: k0,k0+1 ; lanes 16-31: k0+2,k0+3
    v2f a;
    a.x = st[row * 68 + ka];
    a.y = st[row * 68 + ka + 1];
#pragma unroll
    for (int t = 0; t < 4; ++t) {            // 4 independent N-tiles pipeline the WMMAs
      const int n = t * 16 + row;
      v2f bb;
      bb.x = W[n * wstride + ka];
      bb.y = W[n * wstride + ka + 1];
      acc[t] = wmma4(a, bb, acc[t]);
    }
  }
}

// BN(scale/shift) [+ GELU] on D-layout accumulators, store transposed to A-layout LDS tile.
__device__ __forceinline__ void epilogue(float* st, const float* sS, int base,
                                         const v8f acc[4], int lane, bool do_gelu) {
  const int nloc  = lane & 15;
  const int mbase = (lane >> 4) << 3;        // D layout: vgpr r -> M=r (lanes 0-15), M=r+8 (lanes 16-31)
#pragma unroll
  for (int t = 0; t < 4; ++t) {
    const int ch = t * 16 + nloc;
    const float s  = sS[base + ch];
    const float sh = sS[base + 64 + ch];
#pragma unroll
    for (int r = 0; r < 8; ++r) {
      float v = acc[t][r] * s + sh;
      if (do_gelu) v = gelu_exact(v);
      st[(mbase + r) * 68 + ch] = v;
    }
  }
}

__launch_bounds__(128)
__global__ void pairembed_fused_kernel(
    const float* __restrict__ x,  const float* __restrict__ z,
    const float* __restrict__ bn0g, const float* __restrict__ bn0b,
    const float* __restrict__ bn0m, const float* __restrict__ bn0v,
    const float* __restrict__ bn1g, const float* __restrict__ bn1b,
    const float* __restrict__ bn1m, const float* __restrict__ bn1v,
    const float* __restrict__ bn2g, const float* __restrict__ bn2b,
    const float* __restrict__ bn2m, const float* __restrict__ bn2v,
    const float* __restrict__ bn3g, const float* __restrict__ bn3b,
    const float* __restrict__ bn3m, const float* __restrict__ bn3v,
    const float* __restrict__ w1, const float* __restrict__ b1,
    const float* __restrict__ w2, const float* __restrict__ b2,
    const float* __restrict__ w3, const float* __restrict__ b3,
    float* __restrict__ out)
{
  // Padded strides chosen so WMMA operand reads are LDS bank-conflict free:
  // bank = (stride*n + k) mod 64 ; stride 68 -> (4n+k) mod 64, stride 20 -> (20n+k) mod 64.
  __shared__ float sW1[64 * 20];       // W1 padded 14 -> 16(+4) cols
  __shared__ float sW2[64 * 68];
  __shared__ float sW3[64 * 68];
  __shared__ float sS[6 * 64];         // s1,t1 | s2,t2 | s3,t3 (BN+bias folded)
  __shared__ float sBN0[2 * 16];       // s0,t0 for the 14 input channels
  __shared__ float sStage[4][16 * 68]; // per-wave activation tile [pair][ch]

  const int tid = threadIdx.x;

  // ---- cooperative weight / BN preload ----
  for (int idx = tid; idx < 64 * 20; idx += 128) {
    const int r = idx / 20, c = idx % 20;
    sW1[idx] = (c < 14) ? w1[r * 14 + c] : 0.0f;
  }
  for (int idx = tid; idx < 64 * 68; idx += 128) {
    const int r = idx / 68, c = idx % 68;
    sW2[idx] = (c < 64) ? w2[r * 64 + c] : 0.0f;
    sW3[idx] = (c < 64) ? w3[r * 64 + c] : 0.0f;
  }
  if (tid < 64) {
    const float s1 = bn1g[tid] * rsqrtf(bn1v[tid] + BN_EPS_F);
    sS[tid]        = s1;
    sS[64 + tid]   = (b1[tid] - bn1m[tid]) * s1 + bn1b[tid];
    const float s2 = bn2g[tid] * rsqrtf(bn2v[tid] + BN_EPS_F);
    sS[128 + tid]  = s2;
    sS[192 + tid]  = (b2[tid] - bn2m[tid]) * s2 + bn2b[tid];
    const float s3 = bn3g[tid] * rsqrtf(bn3v[tid] + BN_EPS_F);
    sS[256 + tid]  = s3;
    sS[320 + tid]  = (b3[tid] - bn3m[tid]) * s3 + bn3b[tid];
  }
  if (tid < 16) {
    if (tid < 14) {
      const float s0 = bn0g[tid] * rsqrtf(bn0v[tid] + BN_EPS_F);
      sBN0[tid]      = s0;
      sBN0[16 + tid] = bn0b[tid] - bn0m[tid] * s0;
    } else {
      sBN0[tid] = 0.0f;
      sBN0[16 + tid] = 0.0f;
    }
  }
  __syncthreads();

  const int wave = tid >> 5;
  const int lane = tid & 31;
  const int g    = blockIdx.x * 4 + wave;      // global 16-pair tile id, 33024 total
  const int bidx = g / 516;
  const int pb   = (g % 516) * 16;             // 8256 pairs = 516 exact tiles per batch
  float* st = sStage[wave];

  const int row   = lane & 15;                 // M (A) / N-within-tile (B)
  const int khalf = (lane >> 4) << 1;          // 0 or 2 (f32 WMMA operand K split)

  // ---- stage 0: pairwise features + BN0 -> st[pair][0..15] ----
  if (lane < 16) {
    const int p = pb + lane;
    int ii, jj;
    decode_ij(p, ii, jj);
    const float* xb = x + (size_t)bidx * 4 * 128;
    const float pxi = xb[ii],       pyi = xb[128 + ii];
    const float pzi = xb[256 + ii], eei = xb[384 + ii];
    const float pxj = xb[jj],       pyj = xb[128 + jj];
    const float pzj = xb[256 + jj], eej = xb[384 + jj];

    const float pti  = sqrtf(fmaxf(pxi * pxi + pyi * pyi, EPS_F));
    const float ptj  = sqrtf(fmaxf(pxj * pxj + pyj * pyj, EPS_F));
    const float rapi = 0.5f * logf(1.0f + 2.0f * pzi / fmaxf(eei - pzi, 1e-20f));
    const float rapj = 0.5f * logf(1.0f + 2.0f * pzj / fmaxf(eej - pzj, 1e-20f));
    const float phii = atan2f(pyi, pxi);
    const float phij = atan2f(pyj, pxj);

    const float PI_F   = 3.14159265358979f;
    const float TWO_PI = 6.28318530717959f;
    float dphi = fmodf(phii - phij + PI_F, TWO_PI);
    if (dphi < 0.0f) dphi += TWO_PI;            // jnp.mod semantics
    dphi -= PI_F;

    const float drap  = rapi - rapj;
    const float delta = sqrtf(drap * drap + dphi * dphi);
    const float ptmin = fminf(pti, ptj);
    const float sx = pxi + pxj, sy = pyi + pyj, sz = pzi + pzj, se = eei + eej;

    float f[14];
    f[0] = logf(fmaxf(delta, EPS_F));
    f[1] = logf(fmaxf(ptmin * delta, EPS_F));
    f[2] = logf(fmaxf(ptmin / fmaxf(pti + ptj, EPS_F), EPS_F));
    f[3] = logf(fmaxf(se * se - sx * sx - sy * sy - sz * sz, EPS_F));

    const float* zb = z + (size_t)bidx * 17 * 128;
#pragma unroll
    for (int q = 0; q < 6; ++q)
      f[4 + q] = zb[(5 + q) * 128 + ii] + zb[(5 + q) * 128 + jj];
    const float d0i = zb[11 * 128 + ii], d1i = zb[13 * 128 + ii];
    const float e0i = zb[12 * 128 + ii], e1i = zb[14 * 128 + ii];
    const float d0j = zb[11 * 128 + jj], d1j = zb[13 * 128 + jj];
    const float e0j = zb[12 * 128 + jj], e1j = zb[14 * 128 + jj];
    f[10] = (d0i + d0j) / (1.0f + d0i * d0j + EPS_F);
    f[11] = (d1i + d1j) / (1.0f + d1i * d1j + EPS_F);
    f[12] = sqrtf(e0i * e0i + e0j * e0j);
    f[13] = sqrtf(e1i * e1i + e1j * e1j);

#pragma unroll
    for (int c = 0; c < 14; ++c)
      st[lane * 68 + c] = f[c] * sBN0[c] + sBN0[16 + c];
    st[lane * 68 + 14] = 0.0f;
    st[lane * 68 + 15] = 0.0f;
  }
  __builtin_amdgcn_wave_barrier();
  asm volatile("s_wait_dscnt 0" ::: "memory");   // per-wave LDS ops are in-order; fence compiler + counter

  const v8f vzero = {0.f, 0.f, 0.f, 0.f, 0.f, 0.f, 0.f, 0.f};
  v8f acc[4];

  // ---- layer 1: 14(pad16) -> 64, GELU ----
#pragma unroll
  for (int t = 0; t < 4; ++t) acc[t] = vzero;
  gemm_tile(st, sW1, 20, 16, row, khalf, acc);
  __builtin_amdgcn_wave_barrier();
  epilogue(st, sS, 0, acc, lane, true);
  __builtin_amdgcn_wave_barrier();

  // ---- layer 2: 64 -> 64, GELU ----
#pragma unroll
  for (int t = 0; t < 4; ++t) acc[t] = vzero;
  gemm_tile(st, sW2, 68, 64, row, khalf, acc);
  __builtin_amdgcn_wave_barrier();
  epilogue(st, sS, 128, acc, lane, true);
  __builtin_amdgcn_wave_barrier();

  // ---- layer 3: 64 -> 64, BN only ----
#pragma unroll
  for (int t = 0; t < 4; ++t) acc[t] = vzero;
  gemm_tile(st, sW3, 68, 64, row, khalf, acc);
  __builtin_amdgcn_wave_barrier();
  epilogue(st, sS, 256, acc, lane, false);
  __builtin_amdgcn_wave_barrier();
  asm volatile("s_wait_dscnt 0" ::: "memory");

  // ---- symmetric scatter: lane handles (pair = lane&15, 32-channel half) ----
  {
    const int p0     = lane & 15;
    const int chbase = (lane >> 4) * 32;
    int ii, jj;
    decode_ij(pb + p0, ii, jj);
    float* yb = out + (size_t)bidx * 64 * 128 * 128;
#pragma unroll 4
    for (int cc = 0; cc < 32; ++cc) {
      const int ch = chbase + cc;
      const float v = st[p0 * 68 + ch];
      yb[((size_t)ch * 128 + ii) * 128 + jj] = v;  // lanes 0-15: consecutive j -> coalesced
      yb[((size_t)ch * 128 + jj) * 128 + ii] = v;  // mirror
    }
  }
}

extern "C" void kernel_launch(void* const* d_in, const int* in_sizes, int n_in,
                              void* d_out, int out_size, void* d_ws, size_t ws_size,
                              hipStream_t stream) {
  // setup_inputs() dict order:
  // 0:x 1:z | 2-5:bn0 g,b,m,v | 6-9:bn1 | 10-13:bn2 | 14-17:bn3 | 18:w1 19:b1 20:w2 21:b2 22:w3 23:b3
  const float* x    = (const float*)d_in[0];
  const float* z    = (const float*)d_in[1];
  const float* bn0g = (const float*)d_in[2];
  const float* bn0b = (const float*)d_in[3];
  const float* bn0m = (const float*)d_in[4];
  const float* bn0v = (const float*)d_in[5];
  const float* bn1g = (const float*)d_in[6];
  const float* bn1b = (const float*)d_in[7];
  const float* bn1m = (const float*)d_in[8];
  const float* bn1v = (const float*)d_in[9];
  const float* bn2g = (const float*)d_in[10];
  const float* bn2b = (const float*)d_in[11];
  const float* bn2m = (const float*)d_in[12];
  const float* bn2v = (const float*)d_in[13];
  const float* bn3g = (const float*)d_in[14];
  const float* bn3b = (const float*)d_in[15];
  const float* bn3m = (const float*)d_in[16];
  const float* bn3v = (const float*)d_in[17];
  const float* w1   = (const float*)d_in[18];
  const float* b1   = (const float*)d_in[19];
  const float* w2   = (const float*)d_in[20];
  const float* b2   = (const float*)d_in[21];
  const float* w3   = (const float*)d_in[22];
  const float* b3   = (const float*)d_in[23];
  float* out = (float*)d_out;

  // 33024 wave-tiles (64 batches x 516 tiles), 4 waves per 128-thread block
  pairembed_fused_kernel<<<dim3(8256), dim3(128), 0, stream>>>(
      x, z, bn0g, bn0b, bn0m, bn0v, bn1g, bn1b, bn1m, bn1v,
      bn2g, bn2b, bn2m, bn2v, bn3g, bn3b, bn3m, bn3v,
      w1, b1, w2, b2, w3, b3, out);
}